// DifferentiableProjectionLayer_13881334300861
// MI455X (gfx1250) — compile-verified
//
#include <hip/hip_runtime.h>

#define NA   64     // assets per row
#define NC   16     // float4 chunks per row
#define NG   8      // groups
#define EPSV 1e-9f

typedef float v8f __attribute__((ext_vector_type(8)));
typedef float v2f __attribute__((ext_vector_type(2)));

__device__ __forceinline__ float clampf(float v, float lo, float hi) {
#if __has_builtin(__builtin_amdgcn_fmed3f)
    return __builtin_amdgcn_fmed3f(v, lo, hi);   // 1 VALU op: v_med3
#else
    return fminf(fmaxf(v, lo), hi);
#endif
}

__global__ __launch_bounds__(256, 1)
void dykstra_proj_kernel(const float* __restrict__ x,
                         const float* __restrict__ box_lb,
                         const float* __restrict__ box_ub,
                         const float* __restrict__ gA,
                         const float* __restrict__ glb,
                         const float* __restrict__ gub,
                         const int*   __restrict__ niter_p,
                         float* __restrict__ out,
                         float* __restrict__ gram_ws,
                         int rows)
{
    __shared__ float4 s_lb4[NC];
    __shared__ float4 s_ub4[NC];
    __shared__ float  s_ns[NG];
    __shared__ float  s_inv[NG];

    const int tid = threadIdx.x;

    // Stage per-asset box bounds into LDS (broadcast-read every iteration).
    if (tid < NC) {
        s_lb4[tid] = ((const float4*)box_lb)[tid];
        s_ub4[tid] = ((const float4*)box_ub)[tid];
    }
    // norm_sq_g = sum_i A[g][i]^2 (general over A), threads 0..7.
    if (tid < NG) {
        float ns = 0.0f;
        for (int i = 0; i < NA; ++i) {
            float a = gA[tid * NA + i];
            ns = fmaf(a, a, ns);
        }
        s_ns[tid]  = ns;
        s_inv[tid] = 1.0f / (ns + EPSV);
    }

#if __has_builtin(__builtin_amdgcn_wmma_f32_16x16x4_f32)
    // Gram matrix A (8x64, zero-padded to 16) times A^T via f32 WMMA.
    // For each K-chunk of 4, the B-fragment of A^T is laid out identically to
    // the A-fragment of A (lane m<16 holds K=0,1; lane m+16 holds K=2,3), so
    // the same register pair feeds both operands. diag = norm_sq.
    if (blockIdx.x == 0 && tid < 32) {       // one full wave: EXEC all-ones
        const int  lane  = tid;
        const int  m     = lane & 15;
        const int  mm    = m & 7;            // clamp row index for safe loads
        const int  koff  = (lane < 16) ? 0 : 2;
        const bool valid = (m < NG);
        v8f acc = {0.f, 0.f, 0.f, 0.f, 0.f, 0.f, 0.f, 0.f};
        #pragma unroll
        for (int k = 0; k < 16; ++k) {
            float a0 = gA[mm * NA + 4 * k + koff];
            float a1 = gA[mm * NA + 4 * k + koff + 1];
            v2f av;
            av.x = valid ? a0 : 0.0f;
            av.y = valid ? a1 : 0.0f;
            acc = __builtin_amdgcn_wmma_f32_16x16x4_f32(
                      false, av, false, av, (short)0, acc, false, false);
        }
        if (gram_ws) {
            #pragma unroll
            for (int r = 0; r < 8; ++r) {
                int M = (lane < 16) ? r : (r + 8);
                gram_ws[M * 16 + m] = acc[r];
            }
        }
    }
#endif

    __syncthreads();

    const int row = blockIdx.x * blockDim.x + tid;
    if (row >= rows) return;

    // Whole row in registers: 16 x float4.
    float4 w[NC];
    {
        const float4* xin = (const float4*)(x + (size_t)row * NA);
        #pragma unroll
        for (int k = 0; k < NC; ++k) w[k] = xin[k];
    }

    float nsv[NG], invv[NG], glbv[NG], gubv[NG];
    #pragma unroll
    for (int g = 0; g < NG; ++g) {
        nsv[g]  = s_ns[g];
        invv[g] = s_inv[g];
        glbv[g] = glb[g];
        gubv[g] = gub[g];
    }

    const int niter = niter_p[0];
    float S[NG], du[NG], dl[NG], tg[NG];

    for (int it = 0; it < niter; ++it) {
        // ---- box clip + segmented group sums (asset i -> group i & 7) ----
        #pragma unroll
        for (int g = 0; g < NG; ++g) S[g] = 0.0f;
        #pragma unroll
        for (int k = 0; k < NC; ++k) {
            float4 lb = s_lb4[k];
            float4 ub = s_ub4[k];
            float4 v = w[k];
            v.x = clampf(v.x, lb.x, ub.x);
            v.y = clampf(v.y, lb.y, ub.y);
            v.z = clampf(v.z, lb.z, ub.z);
            v.w = clampf(v.w, lb.w, ub.w);
            w[k] = v;
            const int gb = (k & 1) ? 4 : 0;
            S[gb + 0] += v.x; S[gb + 1] += v.y;
            S[gb + 2] += v.z; S[gb + 3] += v.w;
        }
        // ---- simplex shift: sum(w) = sum of group sums (partition) ----
        float total = ((S[0] + S[1]) + (S[2] + S[3])) +
                      ((S[4] + S[5]) + (S[6] + S[7]));
        float t = (total - 1.0f) * (1.0f / (float)NA);
        // Track group sums analytically: S_g <- S_g - 8t (8 members/group).
        #pragma unroll
        for (int g = 0; g < NG; ++g) S[g] = fmaf(-8.0f, t, S[g]);

        // ---- upper-bound delta (branch-free; groups are disjoint) ----
        #pragma unroll
        for (int g = 0; g < NG; ++g)
            du[g] = fmaxf(S[g] - gubv[g], 0.0f) * invv[g];
        #pragma unroll
        for (int g = 0; g < NG; ++g) S[g] = fmaf(-nsv[g], du[g], S[g]);

        // ---- lower-bound delta on recomputed sums ----
        #pragma unroll
        for (int g = 0; g < NG; ++g)
            dl[g] = fminf(S[g] - glbv[g], 0.0f) * invv[g];

        // ---- single fused per-element update: w -= (t + du_g + dl_g) ----
        #pragma unroll
        for (int g = 0; g < NG; ++g) tg[g] = t + (du[g] + dl[g]);
        #pragma unroll
        for (int k = 0; k < NC; ++k) {
            const int gb = (k & 1) ? 4 : 0;
            w[k].x -= tg[gb + 0]; w[k].y -= tg[gb + 1];
            w[k].z -= tg[gb + 2]; w[k].w -= tg[gb + 3];
        }
    }

    // ---- final box_simplex + store ----
    float tot = 0.0f;
    #pragma unroll
    for (int k = 0; k < NC; ++k) {
        float4 lb = s_lb4[k];
        float4 ub = s_ub4[k];
        float4 v = w[k];
        v.x = clampf(v.x, lb.x, ub.x);
        v.y = clampf(v.y, lb.y, ub.y);
        v.z = clampf(v.z, lb.z, ub.z);
        v.w = clampf(v.w, lb.w, ub.w);
        w[k] = v;
        tot += (v.x + v.y) + (v.z + v.w);
    }
    float tf = (tot - 1.0f) * (1.0f / (float)NA);
    float4* o = (float4*)(out + (size_t)row * NA);
    #pragma unroll
    for (int k = 0; k < NC; ++k) {
        float4 v = w[k];
        v.x -= tf; v.y -= tf; v.z -= tf; v.w -= tf;
        o[k] = v;
    }
}

extern "C" void kernel_launch(void* const* d_in, const int* in_sizes, int n_in,
                              void* d_out, int out_size, void* d_ws, size_t ws_size,
                              hipStream_t stream) {
    const float* x      = (const float*)d_in[0];
    const float* box_lb = (const float*)d_in[1];
    const float* box_ub = (const float*)d_in[2];
    const float* gA     = (const float*)d_in[3];
    const float* glb    = (const float*)d_in[4];
    const float* gub    = (const float*)d_in[5];
    const int*   niter  = (const int*)d_in[6];
    float* out = (float*)d_out;
    float* gram = (ws_size >= 256 * sizeof(float)) ? (float*)d_ws : nullptr;

    const int rows = in_sizes[0] / NA;                 // 262144
    dim3 block(256);
    dim3 grid((rows + 255) / 256);
    hipLaunchKernelGGL(dykstra_proj_kernel, grid, block, 0, stream,
                       x, box_lb, box_ub, gA, glb, gub, niter, out, gram, rows);
}